// CF_68169720922624
// MI455X (gfx1250) — compile-verified
//
#include <hip/hip_runtime.h>
#include <math.h>

// Problem dims (match reference setup_inputs)
#define U_DIM 8001
#define I_DIM 16001
#define B_DIM 8192
// split-K config: NK chunks of CS u's each; CS multiple of 4 (WMMA K step)
#define NK 16
#define CS 504   // 16*504 = 8064 >= 8001

typedef __attribute__((ext_vector_type(2))) float v2f;
typedef __attribute__((ext_vector_type(8))) float v8f;

// ---------------------------------------------------------------------------
// Stage 1: per-user bias = sum(nonzero)/count(nonzero) over each rating row.
// One 256-thread block per row, coalesced streaming read (512 MB total).
// ---------------------------------------------------------------------------
__global__ void cf_user_bias_kernel(const float* __restrict__ rat,
                                    float* __restrict__ ub) {
  const int row = blockIdx.x;
  const float* r = rat + (size_t)row * I_DIM;
  float s = 0.f, c = 0.f;
  for (int j = threadIdx.x; j < I_DIM; j += 256) {
    float v = r[j];
    if (v != 0.f) { s += v; c += 1.f; }
  }
  __shared__ float sh_s[256];
  __shared__ float sh_c[256];
  const int t = threadIdx.x;
  sh_s[t] = s; sh_c[t] = c;
  __syncthreads();
  for (int off = 128; off > 0; off >>= 1) {
    if (t < off) { sh_s[t] += sh_s[t + off]; sh_c[t] += sh_c[t + off]; }
    __syncthreads();
  }
  if (t == 0) {
    float S = sh_s[0], C = sh_c[0];
    ub[row] = (C > 0.f) ? (S / fmaxf(C, 1.f)) : 0.f;
  }
}

// ---------------------------------------------------------------------------
// Stage 2: score[b] partial sums via V_WMMA_F32_16X16X4_F32.
// Each wave32 owns 16 b's (M == N == same 16 b's); loops u in steps of 4.
//   A[m][k] = sim[user[b_m], u0+k]          (16x4 f32)
//   B[k][n] = rat[u0+k, item[b_n]] - ub[u0+k]  (4x16 f32)
//   D = A*B + C ; only diag(D) is wanted -> extracted after the K loop.
// Lane layout per ISA 7.12.2 (wave32):
//   A/B: lanes 0-15 carry M/N=0..15 with K={0,1}; lanes 16-31 carry K={2,3}.
//   C/D: VGPR r holds M=r (lanes 0-15, N=lane) and M=r+8 (lanes 16-31, N=lane-16).
// Grid: x = B/128 (8 waves per block * 16 b's), y = NK split-K chunks.
// ---------------------------------------------------------------------------
__global__ void cf_score_wmma_kernel(const int* __restrict__ user,
                                     const int* __restrict__ item,
                                     const float* __restrict__ rat,
                                     const float* __restrict__ sim,
                                     const float* __restrict__ ub,
                                     float* __restrict__ partial) {
  const int lane  = threadIdx.x & 31;
  const int wave  = threadIdx.x >> 5;
  const int m     = lane & 15;     // this lane's b within the 16-group (M and N)
  const int hi    = lane >> 4;     // 0: K={0,1}, 1: K={2,3}
  const int bbase = blockIdx.x * 128 + wave * 16;
  const int b     = bbase + m;
  const int chunk = blockIdx.y;

  const int u_beg = chunk * CS;
  const int u_end = (u_beg + CS < U_DIM) ? (u_beg + CS) : U_DIM;

  const float* __restrict__ srow = sim + (size_t)user[b] * U_DIM;
  const float* __restrict__ rcol = rat + item[b];

  v8f acc = {};
  int u0 = u_beg;
  // fast loop: full K=4 groups
  for (; u0 + 4 <= u_end; u0 += 4) {
    const int ua = u0 + 2 * hi;
    v2f a, bb;
    a.x  = srow[ua];
    a.y  = srow[ua + 1];
    bb.x = rcol[(size_t)ua * I_DIM]       - ub[ua];
    bb.y = rcol[(size_t)(ua + 1) * I_DIM] - ub[ua + 1];
    acc = __builtin_amdgcn_wmma_f32_16x16x4_f32(false, a, false, bb,
                                                (short)0, acc, false, false);
  }
  // tail (U_DIM = 8001 -> last chunk has a 1-element remainder): zero-pad K
  if (u0 < u_end) {
    const int rem = u_end - u0;
    const int k0 = 2 * hi, k1 = 2 * hi + 1;
    v2f a = {0.f, 0.f}, bb = {0.f, 0.f};
    if (k0 < rem) {
      a.x  = srow[u0 + k0];
      bb.x = rcol[(size_t)(u0 + k0) * I_DIM] - ub[u0 + k0];
    }
    if (k1 < rem) {
      a.y  = srow[u0 + k1];
      bb.y = rcol[(size_t)(u0 + k1) * I_DIM] - ub[u0 + k1];
    }
    acc = __builtin_amdgcn_wmma_f32_16x16x4_f32(false, a, false, bb,
                                                (short)0, acc, false, false);
  }

  // diagonal extraction: (m,m) lives in acc[m&7] on lane m (m<8) / lane m+16 (m>=8)
  float* __restrict__ pout = partial + (size_t)chunk * B_DIM + bbase;
#pragma unroll
  for (int r = 0; r < 8; ++r) {
    if (lane == r)      pout[r]     = acc[r];   // M = r,   N = lane   = r
    if (lane == r + 24) pout[r + 8] = acc[r];   // M = r+8, N = lane-16 = r+8
  }
}

// ---------------------------------------------------------------------------
// Stage 3: deterministic split-K reduction + biases + sigmoid*5.
// ---------------------------------------------------------------------------
__global__ void cf_final_kernel(const int* __restrict__ user,
                                const int* __restrict__ item,
                                const float* __restrict__ ubias,
                                const float* __restrict__ ibias,
                                const float* __restrict__ gbias,
                                const float* __restrict__ partial,
                                float* __restrict__ out) {
  const int b = blockIdx.x * 256 + threadIdx.x;
  if (b >= B_DIM) return;
  float s = 0.f;
#pragma unroll
  for (int k = 0; k < NK; ++k) s += partial[(size_t)k * B_DIM + b];
  const float pred = s + ubias[user[b]] + ibias[item[b]] + gbias[0];
  out[b] = 5.f / (1.f + __expf(-pred));
}

// ---------------------------------------------------------------------------
extern "C" void kernel_launch(void* const* d_in, const int* in_sizes, int n_in,
                              void* d_out, int out_size, void* d_ws, size_t ws_size,
                              hipStream_t stream) {
  const int*   user  = (const int*)d_in[0];
  const int*   item  = (const int*)d_in[1];
  const float* rat   = (const float*)d_in[2];
  const float* sim   = (const float*)d_in[3];
  const float* ubias = (const float*)d_in[4];
  const float* ibias = (const float*)d_in[5];
  const float* gbias = (const float*)d_in[6];
  float* out = (float*)d_out;

  // workspace layout: [NK*B partials][U user-bias]
  float* partial = (float*)d_ws;
  float* ub      = partial + (size_t)NK * B_DIM;

  cf_user_bias_kernel<<<U_DIM, 256, 0, stream>>>(rat, ub);

  dim3 g2(B_DIM / 128, NK);
  cf_score_wmma_kernel<<<g2, 256, 0, stream>>>(user, item, rat, sim, ub, partial);

  cf_final_kernel<<<(B_DIM + 255) / 256, 256, 0, stream>>>(user, item, ubias,
                                                           ibias, gbias, partial, out);
}